// GraphConv_62268435857505
// MI455X (gfx1250) — compile-verified
//
#include <hip/hip_runtime.h>

#define C 64
#define N_HOPS 3

typedef float v2f __attribute__((ext_vector_type(2)));
typedef float v8f __attribute__((ext_vector_type(8)));

// ---------------------------------------------------------------------------
// Edge scatter: ent_agg[head] += entity_old[tail] * weight[type-1]; cnt[head]++
// One wave32 per edge, each lane owns 2 channels (float2). Edge id is forced
// into an SGPR so index/type loads become scalar (s_load) and address math SALU.
// ---------------------------------------------------------------------------
__global__ __launch_bounds__(256) void edge_scatter_kernel(
    const float* __restrict__ ent_old, const float* __restrict__ weight,
    const int* __restrict__ head, const int* __restrict__ tail,
    const int* __restrict__ etype, float* __restrict__ agg,
    float* __restrict__ cnt, int E)
{
  int gid  = blockIdx.x * blockDim.x + threadIdx.x;
  int lane = threadIdx.x & 31;
  int e    = __builtin_amdgcn_readfirstlane(gid >> 5);   // wave-uniform -> SGPR
  if (e >= E) return;
  int h = head[e], t = tail[e], ty = etype[e] - 1;
  float2 a = reinterpret_cast<const float2*>(ent_old + (size_t)t * C)[lane];
  float2 b = reinterpret_cast<const float2*>(weight  + (size_t)ty * C)[lane];
  float* dst = agg + (size_t)h * C + lane * 2;
  atomicAdd(dst,     a.x * b.x);
  atomicAdd(dst + 1, a.y * b.y);
  if (lane == 0) atomicAdd(cnt + h, 1.0f);
}

// ---------------------------------------------------------------------------
// Node scatter: node_agg[eh] += all_emb[et] * extra_weight[type]
// all_emb = concat(node_old[:n_users], ent_old). No count needed: the
// scatter-mean divisor is a per-row positive scalar that cancels in l2norm.
// ---------------------------------------------------------------------------
__global__ __launch_bounds__(256) void node_scatter_kernel(
    const float* __restrict__ node_old, const float* __restrict__ ent_old,
    const float* __restrict__ xweight, const int* __restrict__ eh,
    const int* __restrict__ et, const int* __restrict__ xtype,
    float* __restrict__ agg, int E2, int n_users)
{
  int gid  = blockIdx.x * blockDim.x + threadIdx.x;
  int lane = threadIdx.x & 31;
  int e    = __builtin_amdgcn_readfirstlane(gid >> 5);
  if (e >= E2) return;
  int h = eh[e], s = et[e], ty = xtype[e];
  const float* srow = (s < n_users) ? (node_old + (size_t)s * C)
                                    : (ent_old + (size_t)(s - n_users) * C);
  float2 a = reinterpret_cast<const float2*>(srow)[lane];
  float2 b = reinterpret_cast<const float2*>(xweight + (size_t)ty * C)[lane];
  float* dst = agg + (size_t)h * C + lane * 2;
  atomicAdd(dst,     a.x * b.x);
  atomicAdd(dst + 1, a.y * b.y);
}

// ---------------------------------------------------------------------------
// Interaction-matrix scatter: user_sum[row] += (ent_agg[col]/max(cnt,1)) * val
// Entity mean-divide is folded into the per-nnz scale.
// ---------------------------------------------------------------------------
__global__ __launch_bounds__(256) void im_scatter_kernel(
    const float* __restrict__ agg, const float* __restrict__ cnt,
    const float* __restrict__ val, const int* __restrict__ row,
    const int* __restrict__ col, float* __restrict__ user_sum, int NNZ)
{
  int gid  = blockIdx.x * blockDim.x + threadIdx.x;
  int lane = threadIdx.x & 31;
  int e    = __builtin_amdgcn_readfirstlane(gid >> 5);
  if (e >= NNZ) return;
  int r = row[e], c0 = col[e];
  float s = val[e] / fmaxf(cnt[c0], 1.0f);
  float2 x = reinterpret_cast<const float2*>(agg + (size_t)c0 * C)[lane];
  float* dst = user_sum + (size_t)r * C + lane * 2;
  atomicAdd(dst,     x.x * s);
  atomicAdd(dst + 1, x.y * s);
}

// ---------------------------------------------------------------------------
// Row-wise L2 normalize via V_WMMA_F32_16X16X4_F32, 16 rows per wave.
// The same per-lane float2 serves as both the A (16x4) chunk and the
// B (4x16) = A^T chunk, so D = Gram(A_chunk); diag(D) accumulates the per-row
// sum of squares over 16 k-chunks (K=64), all in f32.
//
// Register-pressure strategy: pass 1 loads each chunk, feeds WMMA, and lets it
// die; pass 2 reloads (WGP$-hot) to scale+store. Row index is CLAMPED instead
// of guarding loads -- diag(D) of row m depends only on row m's lanes, so
// clamped duplicate rows cannot corrupt valid norms; only stores are guarded.
// Loads therefore run at full EXEC (required state for WMMA anyway).
// ---------------------------------------------------------------------------
template<bool WRITE_OUT, bool ACC_RES>
__global__ __launch_bounds__(256) void l2norm_wmma_kernel(
    const float* __restrict__ X, float* __restrict__ out,
    float* __restrict__ res, int nrows)
{
  const int lane  = threadIdx.x & 31;
  const int gwave = blockIdx.x * (blockDim.x >> 5) + (threadIdx.x >> 5);
  const int r0    = gwave * 16;
  if (r0 >= nrows) return;                      // wave-uniform exit

  const int rloc = lane & 15;
  const int row  = r0 + rloc;
  const int rowc = (row < nrows) ? row : (nrows - 1);   // clamp, don't diverge
  const int koff = (lane < 16) ? 0 : 2;         // lanes 16-31 hold K=2,3
  const float* rp = X + (size_t)rowc * C;

  // Pass 1: WMMA Gram accumulation (chunks die immediately after use).
  v8f acc = {};
#pragma unroll
  for (int c = 0; c < 16; ++c) {
    v2f a = *reinterpret_cast<const v2f*>(rp + 4 * c + koff);
    acc = __builtin_amdgcn_wmma_f32_16x16x4_f32(
        false, a, false, a, (short)0, acc, false, false);
  }

  // Diagonal D[m,m]: m<8 -> lane m, acc[m];  m>=8 -> lane m+16, acc[m-8]
  int idx = (lane < 8) ? lane : ((lane >= 24) ? (lane - 24) : -1);
  float d = 0.0f;
#pragma unroll
  for (int i = 0; i < 8; ++i) d = (idx == i) ? acc[i] : d;
  int srcLane = (rloc < 8) ? rloc : (rloc + 16);
  float sumsq = __shfl(d, srcLane, 32);
  float scale = 1.0f / fmaxf(sqrtf(sumsq), 1e-12f);

  // Keep pass-2 loads as real reloads (cache-hot) instead of 32 live VGPRs.
  asm volatile("" ::: "memory");

  // Pass 2: reload, scale, store (guarded per-lane; clamped rows write nothing).
  if (row < nrows) {
#pragma unroll
    for (int c = 0; c < 16; ++c) {
      v2f a = *reinterpret_cast<const v2f*>(rp + 4 * c + koff);
      float2 y; y.x = a[0] * scale; y.y = a[1] * scale;
      size_t off = (size_t)row * C + 4 * c + koff;
      if (WRITE_OUT) *reinterpret_cast<float2*>(out + off) = y;
      if (ACC_RES) {
        float2 r2 = *reinterpret_cast<float2*>(res + off);
        r2.x += y.x; r2.y += y.y;
        *reinterpret_cast<float2*>(res + off) = r2;
      }
    }
  }
}

// ---------------------------------------------------------------------------
extern "C" void kernel_launch(void* const* d_in, const int* in_sizes, int n_in,
                              void* d_out, int out_size, void* d_ws, size_t ws_size,
                              hipStream_t stream) {
  const float* user_in  = (const float*)d_in[0];
  const float* ent_in   = (const float*)d_in[1];
  const float* weight   = (const float*)d_in[2];
  const float* xweight  = (const float*)d_in[3];
  const float* im_val   = (const float*)d_in[4];
  const int*   edge_idx = (const int*)d_in[5];
  const int*   edge_ty  = (const int*)d_in[6];
  const int*   xedge_idx= (const int*)d_in[7];
  const int*   xedge_ty = (const int*)d_in[8];
  const int*   im_row   = (const int*)d_in[9];
  const int*   im_col   = (const int*)d_in[10];

  const int n_users = in_sizes[0] / C;
  const int n_ent   = in_sizes[1] / C;
  const int n_nodes = n_users + n_ent;
  const int E   = in_sizes[6];
  const int E2  = in_sizes[8];
  const int NNZ = in_sizes[4];

  const int* head = edge_idx;          const int* tail = edge_idx + E;
  const int* eh   = xedge_idx;         const int* et   = xedge_idx + E2;

  // Workspace layout (floats). Zero-init region is contiguous at the front.
  float* ws = (float*)d_ws;
  size_t o = 0;
  float* ent_agg    = ws + o; o += (size_t)n_ent   * C;
  float* node_agg   = ws + o; o += (size_t)n_nodes * C;
  float* user_sum   = ws + o; o += (size_t)n_users * C;
  float* ent_cnt    = ws + o; o += (size_t)n_ent;
  const size_t zero_bytes = o * sizeof(float);
  float* entity_emb = ws + o; o += (size_t)n_ent   * C;
  float* node_emb   = ws + o;

  // Output layout: [user_res | entity_emb_final | node_res]
  float* out      = (float*)d_out;
  float* user_res = out;                            // n_users * C
  float* out_ent  = out + (size_t)n_users * C;      // n_ent * C
  float* node_res = out + (size_t)n_nodes * C;      // n_nodes * C

  const size_t uB = (size_t)n_users * C * sizeof(float);
  const size_t eB = (size_t)n_ent   * C * sizeof(float);

  // Deterministic per-launch init (graph-replay safe).
  hipMemcpyAsync(node_emb,                       user_in, uB, hipMemcpyDeviceToDevice, stream);
  hipMemcpyAsync(node_emb + (size_t)n_users * C, ent_in,  eB, hipMemcpyDeviceToDevice, stream);
  hipMemcpyAsync(entity_emb,                     ent_in,  eB, hipMemcpyDeviceToDevice, stream);
  hipMemcpyAsync(user_res,                       user_in, uB, hipMemcpyDeviceToDevice, stream);
  hipMemcpyAsync(node_res,                       user_in, uB, hipMemcpyDeviceToDevice, stream);
  hipMemcpyAsync(node_res + (size_t)n_users * C, ent_in,  eB, hipMemcpyDeviceToDevice, stream);

  const int TPB = 256;                   // 8 wave32s per block
  const int ROWS_PER_BLOCK = 8 * 16;     // 8 waves * 16 rows each

  for (int hop = 0; hop < N_HOPS; ++hop) {
    hipMemsetAsync(d_ws, 0, zero_bytes, stream);

    edge_scatter_kernel<<<(E + 7) / 8, TPB, 0, stream>>>(
        entity_emb, weight, head, tail, edge_ty, ent_agg, ent_cnt, E);

    node_scatter_kernel<<<(E2 + 7) / 8, TPB, 0, stream>>>(
        node_emb, entity_emb, xweight, eh, et, xedge_ty, node_agg, E2, n_users);

    im_scatter_kernel<<<(NNZ + 7) / 8, TPB, 0, stream>>>(
        ent_agg, ent_cnt, im_val, im_row, im_col, user_sum, NNZ);

    // entity_emb <- l2norm(ent_agg)   (mean divisor cancels in l2norm)
    l2norm_wmma_kernel<true, false>
        <<<(n_ent + ROWS_PER_BLOCK - 1) / ROWS_PER_BLOCK, TPB, 0, stream>>>(
            ent_agg, entity_emb, nullptr, n_ent);
    // node_emb <- l2norm(node_agg); node_res += node_emb
    l2norm_wmma_kernel<true, true>
        <<<(n_nodes + ROWS_PER_BLOCK - 1) / ROWS_PER_BLOCK, TPB, 0, stream>>>(
            node_agg, node_emb, node_res, n_nodes);
    // user_res += l2norm(user_sum)    (normalized user_emb is otherwise dead)
    l2norm_wmma_kernel<false, true>
        <<<(n_users + ROWS_PER_BLOCK - 1) / ROWS_PER_BLOCK, TPB, 0, stream>>>(
            user_sum, nullptr, user_res, n_users);
  }

  // gcn_res_emb second half = final normalized entity_emb
  hipMemcpyAsync(out_ent, entity_emb, eB, hipMemcpyDeviceToDevice, stream);
}